// MalwareDetector_63204738728607
// MI455X (gfx1250) — compile-verified
//
#include <hip/hip_runtime.h>

typedef __attribute__((ext_vector_type(2))) float v2f;
typedef __attribute__((ext_vector_type(8))) float v8f;

#define F_IN 256
#define G_GRAPHS 128

// ---------------------------------------------------------------------------
// 1) degree histograms via float atomics (resolved at L2 atomic units)
// ---------------------------------------------------------------------------
__global__ void deg_kernel(const int* __restrict__ src, const int* __restrict__ dst,
                           float* __restrict__ out_deg, float* __restrict__ in_deg, int E) {
    int e = blockIdx.x * blockDim.x + threadIdx.x;
    if (e >= E) return;
    atomicAdd(&out_deg[src[e]], 1.0f);
    atomicAdd(&in_deg[dst[e]], 1.0f);
}

// ---------------------------------------------------------------------------
// 2) norms: clamp(deg,1)^-0.5
// ---------------------------------------------------------------------------
__global__ void norm_kernel(const float* __restrict__ out_deg, const float* __restrict__ in_deg,
                            float* __restrict__ srcn, float* __restrict__ dstn, int n) {
    int i = blockIdx.x * blockDim.x + threadIdx.x;
    if (i >= n) return;
    srcn[i] = rsqrtf(fmaxf(out_deg[i], 1.0f));
    dstn[i] = rsqrtf(fmaxf(in_deg[i], 1.0f));
}

// ---------------------------------------------------------------------------
// 3) fused (prev-layer epilogue + scale) @ W  via V_WMMA_F32_16X16X4_F32
//    One wave = one 16-row node block, all NOUT/16 column tiles (A reuse).
//    A fragment (fp32 16x4): lanes 0-15 hold {K,K+1} of row M=lane,
//                            lanes 16-31 hold {K+2,K+3} of row M=lane-16.
//    B fragment (fp32 4x16): VGPR0 = row K (lanes 0-15) / K+2 (16-31),
//                            VGPR1 = row K+1 / K+3, N = lane%16.
//    C/D: VGPR v, lanes 0-15 -> (M=v, N=lane); lanes 16-31 -> (M=v+8).
// ---------------------------------------------------------------------------
template<int K, int NOUT, bool FUSE_PRE>
__global__ __launch_bounds__(256)
void gemm_wmma(const float* __restrict__ A_raw,   // [n, K]  features or agg of prev layer
               const float* __restrict__ W,       // [K, NOUT]
               const float* __restrict__ srcn,    // [n]
               const float* __restrict__ dstn,    // [n]  (FUSE_PRE only)
               const float* __restrict__ bias,    // [K]  (FUSE_PRE only: prev-layer bias)
               float* __restrict__ Out,           // [n, NOUT]
               int n) {
    constexpr int NT = NOUT / 16;
    int wave = (blockIdx.x * blockDim.x + threadIdx.x) >> 5;
    int lane = threadIdx.x & 31;
    int row_base = wave * 16;
    if (row_base >= n) return;          // wave-uniform; EXEC stays all-ones

    int m    = lane & 15;
    int half = lane >> 4;               // 0 or 1
    int row  = row_base + m;

    float sn = srcn[row];
    float dn = FUSE_PRE ? dstn[row] : 0.0f;

    v8f acc[NT] = {};
    const float* arow = A_raw + (size_t)row * K;

    for (int k0 = 0; k0 < K; k0 += 4) {
        int kk = k0 + half * 2;
        float a0 = arow[kk + 0];
        float a1 = arow[kk + 1];
        if (FUSE_PRE) {                 // relu(agg*dst_norm + b_prev) fused in
            a0 = fmaxf(fmaf(a0, dn, bias[kk + 0]), 0.0f);
            a1 = fmaxf(fmaf(a1, dn, bias[kk + 1]), 0.0f);
        }
        v2f a;
        a.x = a0 * sn;
        a.y = a1 * sn;
#pragma unroll
        for (int t = 0; t < NT; ++t) {
            int col = t * 16 + m;
            v2f b;
            b.x = W[(size_t)(kk + 0) * NOUT + col];
            b.y = W[(size_t)(kk + 1) * NOUT + col];
            acc[t] = __builtin_amdgcn_wmma_f32_16x16x4_f32(
                /*neg_a=*/false, a, /*neg_b=*/false, b,
                /*c_mod=*/(short)0, acc[t], /*reuse_a=*/false, /*reuse_b=*/false);
        }
    }
#pragma unroll
    for (int t = 0; t < NT; ++t)
#pragma unroll
        for (int v = 0; v < 8; ++v)
            Out[(size_t)(row_base + v + half * 8) * NOUT + t * 16 + m] = acc[t][v];
}

// ---------------------------------------------------------------------------
// 4) edge gather -> scatter-add.  float4 gather per thread (coalesced rows,
//    L2-resident), 4 float atomics to the destination row.
// ---------------------------------------------------------------------------
template<int F>
__global__ void edge_agg(const float* __restrict__ hw, const int* __restrict__ src,
                         const int* __restrict__ dst, float* __restrict__ agg, int E) {
    constexpr int C = F / 4;
    long tid = (long)blockIdx.x * blockDim.x + threadIdx.x;
    if (tid >= (long)E * C) return;
    int e = (int)(tid / C);
    int c = (int)(tid % C);
    const float4 v = *(const float4*)(hw + (size_t)src[e] * F + c * 4);
    float* p = agg + (size_t)dst[e] * F + c * 4;
    atomicAdd(p + 0, v.x);
    atomicAdd(p + 1, v.y);
    atomicAdd(p + 2, v.z);
    atomicAdd(p + 3, v.w);
}

// ---------------------------------------------------------------------------
// 5) last-layer epilogue fused with graph mean-pool accumulation
// ---------------------------------------------------------------------------
__global__ void pool_kernel(const float* __restrict__ agg2, const float* __restrict__ dstn,
                            const float* __restrict__ b2, const int* __restrict__ gid,
                            float* __restrict__ sums, float* __restrict__ cnt, int n) {
    int i = blockIdx.x * blockDim.x + threadIdx.x;
    if (i >= n) return;
    int g = gid[i];
    float dn = dstn[i];
#pragma unroll
    for (int f = 0; f < 16; ++f) {
        float h = fmaxf(fmaf(agg2[(size_t)i * 16 + f], dn, b2[f]), 0.0f);
        atomicAdd(&sums[g * 16 + f], h);
    }
    atomicAdd(&cnt[g], 1.0f);
}

// ---------------------------------------------------------------------------
// 6) head: logits[g] = mean(h_g) . Wc + bc
// ---------------------------------------------------------------------------
__global__ void final_kernel(const float* __restrict__ sums, const float* __restrict__ cnt,
                             const float* __restrict__ Wc, const float* __restrict__ bc,
                             float* __restrict__ out, int g_total) {
    int g = blockIdx.x * blockDim.x + threadIdx.x;
    if (g >= g_total) return;
    float inv = 1.0f / fmaxf(cnt[g], 1.0f);
    float acc = bc[0];
#pragma unroll
    for (int f = 0; f < 16; ++f)
        acc = fmaf(sums[g * 16 + f] * inv, Wc[f], acc);
    out[g] = acc;
}

// ---------------------------------------------------------------------------
extern "C" void kernel_launch(void* const* d_in, const int* in_sizes, int n_in,
                              void* d_out, int out_size, void* d_ws, size_t ws_size,
                              hipStream_t stream) {
    const float* features = (const float*)d_in[0];
    const int*   src      = (const int*)d_in[1];
    const int*   dst      = (const int*)d_in[2];
    const int*   gid      = (const int*)d_in[3];
    const float* W0 = (const float*)d_in[4];
    const float* b0 = (const float*)d_in[5];
    const float* W1 = (const float*)d_in[6];
    const float* b1 = (const float*)d_in[7];
    const float* W2 = (const float*)d_in[8];
    const float* b2 = (const float*)d_in[9];
    const float* Wc = (const float*)d_in[10];
    const float* bc = (const float*)d_in[11];
    float* out = (float*)d_out;

    const int N = in_sizes[0] / F_IN;   // 131072
    const int E = in_sizes[1];          // 2097152
    const int G = out_size;             // 128

    // workspace layout (floats); hw/agg slabs reused across layers
    float* ws       = (float*)d_ws;
    float* out_deg  = ws;                     // N
    float* in_deg   = out_deg + N;            // N
    float* srcn     = in_deg + N;             // N
    float* dstn     = srcn + N;               // N
    float* bufHW    = dstn + N;               // N*64 (hw0 / hw1 / hw2)
    float* bufAGG   = bufHW + (size_t)N * 64; // N*64 (agg0 / agg1 / agg2)
    float* sums     = bufAGG + (size_t)N * 64;// G*16
    float* cnt      = sums + (size_t)G * 16;  // G

    const int TPB = 256;

    // degrees + norms
    hipMemsetAsync(out_deg, 0, (size_t)2 * N * sizeof(float), stream);   // out_deg+in_deg
    deg_kernel<<<(E + TPB - 1) / TPB, TPB, 0, stream>>>(src, dst, out_deg, in_deg, E);
    norm_kernel<<<(N + TPB - 1) / TPB, TPB, 0, stream>>>(out_deg, in_deg, srcn, dstn, N);

    const int gemm_blocks = (N / 16) * 32 / TPB;     // one wave per 16 rows

    // ---- layer 0: hw0 = (x*srcn) @ W0 ; agg0 = scatter_add(hw0[src] -> dst)
    gemm_wmma<256, 64, false><<<gemm_blocks, TPB, 0, stream>>>(
        features, W0, srcn, nullptr, nullptr, bufHW, N);
    hipMemsetAsync(bufAGG, 0, (size_t)N * 64 * sizeof(float), stream);
    {
        long tot = (long)E * 16;
        edge_agg<64><<<(int)((tot + TPB - 1) / TPB), TPB, 0, stream>>>(bufHW, src, dst, bufAGG, E);
    }

    // ---- layer 1: hw1 = (relu(agg0*dstn+b0)*srcn) @ W1 ; agg1
    gemm_wmma<64, 32, true><<<gemm_blocks, TPB, 0, stream>>>(
        bufAGG, W1, srcn, dstn, b0, bufHW, N);
    hipMemsetAsync(bufAGG, 0, (size_t)N * 32 * sizeof(float), stream);
    {
        long tot = (long)E * 8;
        edge_agg<32><<<(int)((tot + TPB - 1) / TPB), TPB, 0, stream>>>(bufHW, src, dst, bufAGG, E);
    }

    // ---- layer 2: hw2 = (relu(agg1*dstn+b1)*srcn) @ W2 ; agg2
    gemm_wmma<32, 16, true><<<gemm_blocks, TPB, 0, stream>>>(
        bufAGG, W2, srcn, dstn, b1, bufHW, N);
    hipMemsetAsync(bufAGG, 0, (size_t)N * 16 * sizeof(float), stream);
    {
        long tot = (long)E * 4;
        edge_agg<16><<<(int)((tot + TPB - 1) / TPB), TPB, 0, stream>>>(bufHW, src, dst, bufAGG, E);
    }

    // ---- pool (fused relu(agg2*dstn+b2)) + head
    hipMemsetAsync(sums, 0, (size_t)(G * 16 + G) * sizeof(float), stream);
    pool_kernel<<<(N + TPB - 1) / TPB, TPB, 0, stream>>>(bufAGG, dstn, b2, gid, sums, cnt, N);
    final_kernel<<<1, G, 0, stream>>>(sums, cnt, Wc, bc, out, G);
}